// VoxDeformCrossAttn_16131897164037
// MI455X (gfx1250) — compile-verified
//
#include <hip/hip_runtime.h>
#include <hip/hip_bf16.h>

// ---------------- CDNA5 fp32 WMMA helpers ----------------
typedef __attribute__((ext_vector_type(2))) float v2f;
typedef __attribute__((ext_vector_type(8))) float v8f;
typedef __attribute__((ext_vector_type(4))) unsigned int v4u;
typedef __attribute__((ext_vector_type(8))) int v8i;
typedef __attribute__((ext_vector_type(4))) int v4i;

__device__ __forceinline__ v8f wmma4(v2f a, v2f b, v8f c) {
  // V_WMMA_F32_16X16X4_F32 : D(16x16,f32) = A(16x4,f32) * B(4x16,f32) + C
  return __builtin_amdgcn_wmma_f32_16x16x4_f32(false, a, false, b, (short)0, c,
                                               false, false);
}

#define LQ      16384
#define HEADS   8
#define POINTS  4
#define LEVELS  3
#define CAMS    2
#define DMODEL  256
#define DIM     32

// ---------------- Tensor Data Mover: 2-D f32 tile global -> LDS ----------------
#ifdef __has_builtin
#if __has_builtin(__builtin_amdgcn_tensor_load_to_lds)
#define HAVE_TDM 1
#endif
#endif
#ifndef HAVE_TDM
#define HAVE_TDM 0
#endif

#if HAVE_TDM
__device__ __forceinline__ unsigned lds_addr_of(const void* p) {
  // generic -> LDS(addrspace 3) cast; AS3 pointers are 32-bit LDS byte offsets
  return (unsigned)(unsigned long long)(const __attribute__((address_space(3))) void*)p;
}

// Load a (tileH x tileW) f32 tile (row stride = strideW elems) at gsrc into LDS,
// packed row-major tileH x tileW. D# built per CDNA5 ISA.
__device__ __forceinline__ void tdm_load_2d_f32(float* ldsDst, const float* gsrc,
                                                unsigned tileW, unsigned tileH,
                                                unsigned tensorW, unsigned tensorH,
                                                unsigned strideW) {
  unsigned long long ga = (unsigned long long)(uintptr_t)gsrc;
  v4u g0;
  g0.x = 1u;                                         // count=1 (valid user D#)
  g0.y = lds_addr_of(ldsDst);                        // lds_addr (bytes)
  g0.z = (unsigned)(ga & 0xFFFFFFFFu);               // global_addr[31:0]
  g0.w = (unsigned)((ga >> 32) & 0x01FFFFFFu)        // global_addr[56:32]
       | 0x80000000u;                                // type=2 ("image"), bits127:126
  v8i g1;
  g1[0] = (int)(2u << 16);                           // data_size=2 -> 4 bytes
  g1[1] = (int)((tensorW & 0xFFFFu) << 16);          // tensor_dim0[15:0]
  g1[2] = (int)((tensorW >> 16) | ((tensorH & 0xFFFFu) << 16));
  g1[3] = (int)((tensorH >> 16) | (tileW << 16));    // tile_dim0
  g1[4] = (int)(tileH & 0xFFFFu);                    // tile_dim1 (tile_dim2=0)
  g1[5] = (int)strideW;                              // tensor_dim0_stride[31:0]
  g1[6] = 0;                                         // stride[47:32]=0, dim1_stride=0
  g1[7] = 0;
  v4i z4 = {0, 0, 0, 0};
#if __clang_major__ >= 23
  v8i z8 = {0, 0, 0, 0, 0, 0, 0, 0};
  __builtin_amdgcn_tensor_load_to_lds(g0, g1, z4, z4, z8, 0);
#else
  __builtin_amdgcn_tensor_load_to_lds(g0, g1, z4, z4, 0);
#endif
}
#endif  // HAVE_TDM

// =====================================================================
// fp32 WMMA GEMM, compile-time MODE (clean inner loops, no flag branches):
//   MODE 0: a = A[row,k]
//   MODE 1: a = A[row,k] + A2[row,k]                        (tgt + query_pos)
//   MODE 2: a = relu(A[row,k]*ascale[k] + ashift[k])        (BN+ReLU on A)
//   MODE 3: a = relu(Abuf[((k>>5)*LQ+row)*32+(k&31)] * ascale[k&31] + ashift[k&31])
//            (head-transpose gather + BN+ReLU)
// Block = 256 threads = 8 waves. Block tile = 128(M) x 16(N):
// wave w -> M-tile blockIdx.x*128 + w*16; N-tile = blockIdx.y*16.
// B slice (K x 16) staged in LDS via TDM, reused by all 8 waves.
// =====================================================================
template <int MODE>
__global__ __launch_bounds__(256)
void gemm_f32_wmma(const float* __restrict__ A, const float* __restrict__ A2,
                   const float* __restrict__ ascale, const float* __restrict__ ashift,
                   const float* __restrict__ B, const float* __restrict__ bias,
                   const float* __restrict__ resid, float* __restrict__ C,
                   int M, int N, int K) {
  __shared__ float sB[256 * 16];          // K_max=256 -> 16 KB
  int lane = threadIdx.x & 31;
  int wave = threadIdx.x >> 5;
  int l15  = lane & 15;
  int kh   = lane >> 4;                   // K-pair half: {0,1} vs {2,3}
  int m0   = blockIdx.x * 128 + wave * 16;
  int n0   = blockIdx.y * 16;
  int row  = m0 + l15;
  int col  = n0 + l15;

#if HAVE_TDM
  if (wave == 0) tdm_load_2d_f32(sB, B + n0, 16u, (unsigned)K, (unsigned)N,
                                 (unsigned)K, (unsigned)N);
#endif

  v8f acc;
  float binit = bias ? bias[col] : 0.0f;
#pragma unroll
  for (int r = 0; r < 8; ++r) acc[r] = binit;

#if HAVE_TDM
  if (wave == 0) __builtin_amdgcn_s_wait_tensorcnt(0);
#else
  for (int i = threadIdx.x; i < K * 16; i += 256)
    sB[i] = B[(unsigned)(i >> 4) * (unsigned)N + (unsigned)(n0 + (i & 15))];
#endif
  __syncthreads();

#pragma unroll 4
  for (int k = 0; k < K; k += 4) {
    int ka = k + 2 * kh;
    v2f a, b;
    if (MODE == 3) {
      int k0 = ka, k1 = ka + 1;
      float a0 = A[(unsigned)((k0 >> 5) * LQ + row) * DIM + (unsigned)(k0 & 31)];
      float a1 = A[(unsigned)((k1 >> 5) * LQ + row) * DIM + (unsigned)(k1 & 31)];
      a.x = fmaxf(a0 * ascale[k0 & 31] + ashift[k0 & 31], 0.0f);
      a.y = fmaxf(a1 * ascale[k1 & 31] + ashift[k1 & 31], 0.0f);
    } else {
      const float2 av = *reinterpret_cast<const float2*>(A + (unsigned)row * (unsigned)K + ka);
      float a0 = av.x, a1 = av.y;
      if (MODE == 1) {
        const float2 av2 = *reinterpret_cast<const float2*>(A2 + (unsigned)row * (unsigned)K + ka);
        a0 += av2.x; a1 += av2.y;
      }
      if (MODE == 2) {
        a0 = fmaxf(a0 * ascale[ka] + ashift[ka], 0.0f);
        a1 = fmaxf(a1 * ascale[ka + 1] + ashift[ka + 1], 0.0f);
      }
      a.x = a0; a.y = a1;
    }
    b.x = sB[ka * 16 + l15];
    b.y = sB[(ka + 1) * 16 + l15];
    acc = wmma4(a, b, acc);
  }

  if (resid) {
#pragma unroll
    for (int r = 0; r < 8; ++r) {
      int rr = m0 + r + 8 * kh;   // C layout: VGPR r -> M=r (lo half) / M=8+r (hi half)
      C[(unsigned)rr * (unsigned)N + col] = acc[r] + resid[(unsigned)rr * (unsigned)N + col];
    }
  } else {
#pragma unroll
    for (int r = 0; r < 8; ++r) {
      int rr = m0 + r + 8 * kh;
      C[(unsigned)rr * (unsigned)N + col] = acc[r];
    }
  }
}

// =====================================================================
// Fused sampling + first level-MLP matmul.
// Block = 128 threads (4 waves), 16 rows per block; W1 staged by TDM in
// parallel with the sampling math. row rid = ((m*Lq + q)*POINTS + p).
// =====================================================================
__global__ __launch_bounds__(128)
void sample_mlp1(const float* __restrict__ val, const float* __restrict__ offlin,
                 const float* __restrict__ refpts, const float* __restrict__ cams,
                 const float* __restrict__ W1, const int* __restrict__ sszp,
                 float* __restrict__ x1pre, int level, int Hc, int Wl, int startRow) {
  __shared__ float sfeat[16 * 64];
  __shared__ float sW1[64 * 64];
  int tid = threadIdx.x;

#if HAVE_TDM
  if ((tid >> 5) == 0) tdm_load_2d_f32(sW1, W1, 64u, 64u, 64u, 64u, 64u);
#endif

  { // ---- Phase A: bilinear sampling; 8 threads per row, 8 dims each ----
    int r    = tid >> 3;
    int sub  = tid & 7;
    int rid  = blockIdx.x * 16 + r;
    int p    = rid & 3;
    int qm   = rid >> 2;            // m*Lq + q
    int q    = qm & (LQ - 1);
    int m    = qm >> 14;
    int t    = sub >> 2;            // camera
    int dsub = (sub & 3) * 8;       // dim sub-block within DIM

    float ss = (float)sszp[0];
    const float* ob = offlin + (unsigned)q * (HEADS * POINTS * 3) + (m * POINTS + p) * 3;
    float o0 = 1.0f / (1.0f + expf(-ob[0]));
    float o1 = 1.0f / (1.0f + expf(-ob[1]));
    float o2 = 1.0f / (1.0f + expf(-ob[2]));
    float lx = refpts[q * 3 + 0] + (o0 * ss - 0.5f * ss);
    float ly = refpts[q * 3 + 1] + (o1 * ss - 0.5f * ss);
    float lz = refpts[q * 3 + 2] + (o2 * ss - 0.5f * ss);

    const float* cm = cams + ((unsigned)level * CAMS + t) * 12;  // (3,4) pinhole
    float u0 = cm[0] * lx + cm[1] * ly + cm[2]  * lz + cm[3];
    float u1 = cm[4] * lx + cm[5] * ly + cm[6]  * lz + cm[7];
    float u2 = cm[8] * lx + cm[9] * ly + cm[10] * lz + cm[11];
    float w  = fmaxf(u2, 1e-5f);
    float sx = u0 / w - 0.5f;       // grid-sample algebra collapses to u-0.5, v-0.5
    float sy = u1 / w - 0.5f;
    float x0f = floorf(sx), y0f = floorf(sy);
    float wx1 = sx - x0f, wy1 = sy - y0f;
    int   x0  = (int)x0f, y0 = (int)y0f;

    float acc[8];
#pragma unroll
    for (int j = 0; j < 8; ++j) acc[j] = 0.0f;
#pragma unroll
    for (int c = 0; c < 4; ++c) {
      int   dy  = c >> 1, dx = c & 1;
      int   yy  = y0 + dy, xx = x0 + dx;
      float wgt = (dy ? wy1 : (1.0f - wy1)) * (dx ? wx1 : (1.0f - wx1));
      bool  ok  = (xx >= 0) && (xx < Wl) && (yy >= 0) && (yy < Hc);
      int   xi  = min(max(xx, 0), Wl - 1);
      int   yi  = min(max(yy, 0), Hc - 1);
      const float* vp = val + ((unsigned)(startRow + (t * Hc + yi) * Wl + xi)) * DMODEL
                            + m * DIM + dsub;
      const float4 va = *reinterpret_cast<const float4*>(vp);
      const float4 vb = *reinterpret_cast<const float4*>(vp + 4);
      float s = ok ? wgt : 0.0f;
      acc[0] += s * va.x; acc[1] += s * va.y; acc[2] += s * va.z; acc[3] += s * va.w;
      acc[4] += s * vb.x; acc[5] += s * vb.y; acc[6] += s * vb.z; acc[7] += s * vb.w;
    }
#pragma unroll
    for (int j = 0; j < 8; ++j) sfeat[r * 64 + t * DIM + dsub + j] = acc[j];
  }

#if HAVE_TDM
  if ((tid >> 5) == 0) __builtin_amdgcn_s_wait_tensorcnt(0);
#else
  for (int i = tid; i < 64 * 64; i += 128) sW1[i] = W1[i];
#endif
  __syncthreads();

  // ---- Phase C: WMMA; wave w -> N-tile w (16 cols), K = 64 ----
  int lane = tid & 31, wave = tid >> 5;
  int l15 = lane & 15, kh = lane >> 4, n0 = wave * 16;
  v8f acc;
#pragma unroll
  for (int r = 0; r < 8; ++r) acc[r] = 0.0f;
#pragma unroll
  for (int k = 0; k < 64; k += 4) {
    int ka = k + 2 * kh;
    v2f a, b;
    a.x = sfeat[l15 * 64 + ka];
    a.y = sfeat[l15 * 64 + ka + 1];
    b.x = sW1[ka * 64 + n0 + l15];
    b.y = sW1[(ka + 1) * 64 + n0 + l15];
    acc = wmma4(a, b, acc);
  }
#pragma unroll
  for (int r = 0; r < 8; ++r) {
    int row = blockIdx.x * 16 + r + 8 * kh;
    x1pre[(unsigned)row * 64 + n0 + l15] = acc[r];
  }
}

// ---------------- BatchNorm statistics (two-stage, deterministic) ----------------
__global__ void col_partials(const float* __restrict__ X, int rows, int cols, int rpb,
                             float* __restrict__ part) {
  int c  = threadIdx.x;              // blockDim.x == cols
  int r0 = blockIdx.x * rpb;
  int r1 = min(r0 + rpb, rows);
  float s = 0.0f, s2 = 0.0f;
  for (int r = r0; r < r1; ++r) {
    float v = X[(size_t)r * cols + c];
    s += v; s2 += v * v;
  }
  part[(size_t)blockIdx.x * 2 * cols + c] = s;
  part[(size_t)blockIdx.x * 2 * cols + cols + c] = s2;
}

__global__ void finalize_stats(const float* __restrict__ part, int nblocks, int cols,
                               float invRows, const float* __restrict__ g,
                               const float* __restrict__ bta,
                               float* __restrict__ scale, float* __restrict__ shift) {
  int c = threadIdx.x;               // blockDim.x == cols, single block
  float s = 0.0f, s2 = 0.0f;
  for (int b = 0; b < nblocks; ++b) {     // fixed order -> bitwise deterministic
    s  += part[(size_t)b * 2 * cols + c];
    s2 += part[(size_t)b * 2 * cols + cols + c];
  }
  float mean = s * invRows;
  float var  = s2 * invRows - mean * mean;   // biased var (matches x.var(0))
  float istd = rsqrtf(var + 1e-5f);
  float sc   = g[c] * istd;
  scale[c] = sc;
  shift[c] = bta[c] - mean * sc;
}

// ---------------- softmax over the 4 points of each (q, head) ----------------
__global__ void softmax4(float* __restrict__ attn, int total) {
  int i = blockIdx.x * blockDim.x + threadIdx.x;
  if (i >= total) return;
  float* p = attn + (size_t)i * 4;
  float a = p[0], b = p[1], c = p[2], d = p[3];
  float mx = fmaxf(fmaxf(a, b), fmaxf(c, d));
  float ea = expf(a - mx), eb = expf(b - mx), ec = expf(c - mx), ed = expf(d - mx);
  float inv = 1.0f / (ea + eb + ec + ed);
  p[0] = ea * inv; p[1] = eb * inv; p[2] = ec * inv; p[3] = ed * inv;
}

// ---------------- BN+ReLU on x2pre then attention-weighted point sum ----------------
__global__ void wsum_points(const float* __restrict__ x2pre, const float* __restrict__ attn,
                            const float* __restrict__ scale, const float* __restrict__ shift,
                            float* __restrict__ fused, int level) {
  int idx = blockIdx.x * blockDim.x + threadIdx.x;   // HEADS*LQ*DIM threads
  int d  = idx & 31;
  int qm = idx >> 5;                  // m*Lq + q
  int q  = qm & (LQ - 1);
  int m  = qm >> 14;
  float sc = scale[d], sh = shift[d];
  float s = 0.0f;
#pragma unroll
  for (int p = 0; p < 4; ++p) {
    float v = x2pre[((size_t)qm * 4 + p) * DIM + d];
    v = fmaxf(v * sc + sh, 0.0f);
    s += v * attn[(size_t)q * 32 + m * 4 + p];
  }
  fused[(size_t)qm * (LEVELS * DIM) + level * DIM + d] = s;
}

// =====================================================================
// Host-side orchestration
// =====================================================================
extern "C" void kernel_launch(void* const* d_in, const int* in_sizes, int n_in,
                              void* d_out, int out_size, void* d_ws, size_t ws_size,
                              hipStream_t stream) {
  (void)in_sizes; (void)n_in; (void)out_size; (void)ws_size;
  // ---- inputs (setup_inputs insertion order; params dict in insertion order) ----
  const float* tgt       = (const float*)d_in[0];
  const float* memory    = (const float*)d_in[1];
  const float* query_pos = (const float*)d_in[2];
  const float* cams      = (const float*)d_in[3];
  const float* refpts    = (const float*)d_in[4];
  const float* W_off     = (const float*)d_in[5];
  const float* b_off     = (const float*)d_in[6];
  const float* W_attn    = (const float*)d_in[7];
  const float* b_attn    = (const float*)d_in[8];
  const float* W_v       = (const float*)d_in[9];
  const float* b_v       = (const float*)d_in[10];
  const float* W_o       = (const float*)d_in[11];
  const float* b_o       = (const float*)d_in[12];
  const float* fd_W1     = (const float*)d_in[13];
  const float* fd_g1     = (const float*)d_in[14];
  const float* fd_b1     = (const float*)d_in[15];
  const float* fd_W2     = (const float*)d_in[16];
  const float* fd_g2     = (const float*)d_in[17];
  const float* fd_b2     = (const float*)d_in[18];
  const float* mW1[3], *mg1[3], *mb1[3], *mW2[3], *mg2[3], *mb2[3];
  for (int l = 0; l < 3; ++l) {
    int b = 19 + l * 6;
    mW1[l] = (const float*)d_in[b + 0];
    mg1[l] = (const float*)d_in[b + 1];
    mb1[l] = (const float*)d_in[b + 2];
    mW2[l] = (const float*)d_in[b + 3];
    mg2[l] = (const float*)d_in[b + 4];
    mb2[l] = (const float*)d_in[b + 5];
  }
  const int* ssz = (const int*)d_in[38];   // sample_size (d_in[37] = spatial_shapes)

  // ---- workspace layout (floats) ----
  float* ws = (float*)d_ws;
  const int S = 21504;                       // 128*128 + 64*64 + 32*32
  float* VAL    = ws;                        // S*256
  float* OFFLIN = VAL    + (size_t)S * 256;  // 16384*96
  float* ATTNW  = OFFLIN + (size_t)LQ * 96;  // 16384*32
  float* FUSED  = ATTNW  + (size_t)LQ * 32;  // 131072*96
  float* BIG1   = FUSED  + (size_t)HEADS * LQ * 96;            // 524288*64
  float* BIG2   = BIG1   + (size_t)HEADS * LQ * POINTS * 64;   // 524288*32
  float* STATS  = BIG2   + (size_t)HEADS * LQ * POINTS * 32;   // 256
  float* PART   = STATS  + 256;              // up to 131,072
  float* SC_A = STATS + 0,  *SH_A = STATS + 96;
  float* SC_B = STATS + 192, *SH_B = STATS + 224;

  const int ROWS1 = HEADS * LQ * POINTS;     // 524,288
  const int ROWS2 = HEADS * LQ;              // 131,072
  static const int Hc_[3]  = {64, 32, 16};
  static const int Wl_[3]  = {128, 64, 32};
  static const int st_[3]  = {0, 16384, 20480};

  // ---- value / offset / attention projections ----
  gemm_f32_wmma<0><<<dim3(S / 128, 16), 256, 0, stream>>>(
      memory, nullptr, nullptr, nullptr, W_v, b_v, nullptr, VAL, S, 256, 256);
  gemm_f32_wmma<0><<<dim3(LQ / 128, 6), 256, 0, stream>>>(
      tgt, nullptr, nullptr, nullptr, W_off, b_off, nullptr, OFFLIN, LQ, 96, 256);
  gemm_f32_wmma<1><<<dim3(LQ / 128, 2), 256, 0, stream>>>(
      tgt, query_pos, nullptr, nullptr, W_attn, b_attn, nullptr, ATTNW, LQ, 32, 256);
  softmax4<<<(LQ * HEADS + 255) / 256, 256, 0, stream>>>(ATTNW, LQ * HEADS);

  // ---- per-level: sample+MLP1, BN stats, MLP2, BN stats, weighted point-sum ----
  for (int l = 0; l < 3; ++l) {
    sample_mlp1<<<ROWS1 / 16, 128, 0, stream>>>(
        VAL, OFFLIN, refpts, cams, mW1[l], ssz, BIG1, l, Hc_[l], Wl_[l], st_[l]);
    col_partials<<<512, 64, 0, stream>>>(BIG1, ROWS1, 64, 1024, PART);
    finalize_stats<<<1, 64, 0, stream>>>(PART, 512, 64, 1.0f / (float)ROWS1,
                                         mg1[l], mb1[l], SC_A, SH_A);
    gemm_f32_wmma<2><<<dim3(ROWS1 / 128, 2), 256, 0, stream>>>(
        BIG1, nullptr, SC_A, SH_A, mW2[l], nullptr, nullptr, BIG2, ROWS1, 32, 64);
    col_partials<<<512, 32, 0, stream>>>(BIG2, ROWS1, 32, 1024, PART);
    finalize_stats<<<1, 32, 0, stream>>>(PART, 512, 32, 1.0f / (float)ROWS1,
                                         mg2[l], mb2[l], SC_B, SH_B);
    wsum_points<<<(HEADS * LQ * DIM) / 256, 256, 0, stream>>>(
        BIG2, ATTNW, SC_B, SH_B, FUSED, l);
  }

  // ---- fuse MLP across levels ----
  gemm_f32_wmma<0><<<dim3(ROWS2 / 128, 6), 256, 0, stream>>>(
      FUSED, nullptr, nullptr, nullptr, fd_W1, nullptr, nullptr, BIG1, ROWS2, 96, 96);
  col_partials<<<256, 96, 0, stream>>>(BIG1, ROWS2, 96, 512, PART);
  finalize_stats<<<1, 96, 0, stream>>>(PART, 256, 96, 1.0f / (float)ROWS2,
                                       fd_g1, fd_b1, SC_A, SH_A);
  gemm_f32_wmma<2><<<dim3(ROWS2 / 128, 2), 256, 0, stream>>>(
      BIG1, nullptr, SC_A, SH_A, fd_W2, nullptr, nullptr, BIG2, ROWS2, 32, 96);
  col_partials<<<256, 32, 0, stream>>>(BIG2, ROWS2, 32, 512, PART);
  finalize_stats<<<1, 32, 0, stream>>>(PART, 256, 32, 1.0f / (float)ROWS2,
                                       fd_g2, fd_b2, SC_B, SH_B);

  // ---- output projection with head-transpose gather + BN + residual ----
  gemm_f32_wmma<3><<<dim3(LQ / 128, 16), 256, 0, stream>>>(
      BIG2, nullptr, SC_B, SH_B, W_o, b_o, tgt, (float*)d_out, LQ, 256, 256);
}